// FNO2d_16432544874758
// MI455X (gfx1250) — compile-verified
//
#include <hip/hip_runtime.h>
#include <hip/hip_bf16.h>

// ---------------------------------------------------------------------------
// FNO2d for MI455X (gfx1250, wave32).
// Spectral path implemented as partial-DFT GEMMs (137 is prime; only 32x16
// modes survive). All GEMM stages use V_WMMA_F32_16X16X4_F32.
// This revision removes ALL guards from the WMMA inner loops:
//   - K padded to a multiple of 4 with zero-filled twiddle rows/cols
//   - out-of-range M/N loads are allowed (row/col independence of WMMA);
//     only stores are guarded. All overreads stay inside d_ws.
// K is a template parameter -> fully unrolled, branch-free inner loops.
// ---------------------------------------------------------------------------

typedef __attribute__((ext_vector_type(2))) float v2f;
typedef __attribute__((ext_vector_type(8))) float v8f;

#define Bn   16
#define Cn   64
#define Sn   128
#define Hh   137              // 128 + PAD(9)
#define HW   18769            // 137*137
#define Mm   16               // retained modes per axis
#define NKX  32               // kx slots: 0..15 and 121..136
#define RY   (Bn*Cn*Hh)       // 140288 rows for y-direction GEMMs (16*8768)
#define NMODE 512             // 32*16

__device__ __forceinline__ float gelu_f(float v) {
  return 0.5f * v * (1.0f + erff(v * 0.70710678118654752f));
}

__device__ __forceinline__ v8f wmma4(v2f a, v2f b, v8f c) {
  // (neg_a, A, neg_b, B, c_mod, C, reuse_a, reuse_b)
  return __builtin_amdgcn_wmma_f32_16x16x4_f32(false, a, false, b, (short)0, c,
                                               false, false);
}

// Unguarded 16x16 tile of C += A(Mx KPAD)*B(KPAD x N); arbitrary strides.
// A frag: lane L -> row tm+(L&15), K pair k+(L>=16?2:0).
// B frag: lane L -> col tn+(L&15), K rows k+(L>=16?2:0), +1.
template<int KPAD>
__device__ __forceinline__ void gemm_tile_u(const float* __restrict__ A, int rsA, int csA,
                                            const float* __restrict__ B, int rsB, int csB,
                                            int tm, int tn, v8f& acc) {
  const int lane = threadIdx.x & 31;
  const int mr = tm + (lane & 15);
  const int nc = tn + (lane & 15);
  const int kh = (lane >> 4) << 1;
  const float* Ap = A + (size_t)mr * rsA + (size_t)kh * csA;
  const float* Bp = B + (size_t)kh * rsB + (size_t)nc * csB;
  #pragma unroll
  for (int k = 0; k < KPAD; k += 4) {
    v2f a, b;
    a.x = Ap[0];  a.y = Ap[csA];
    b.x = Bp[0];  b.y = Bp[rsB];
    acc = wmma4(a, b, acc);
    Ap += (size_t)4 * csA;
    Bp += (size_t)4 * rsB;
  }
}

// Unguarded complex tile: (Ar+iAi)(Br+iBi) -> cre, cim. 3 accumulators.
template<int KPAD>
__device__ __forceinline__ void cgemm_tile_u(const float* __restrict__ Ar, const float* __restrict__ Ai,
                                             int rsA, int csA,
                                             const float* __restrict__ Br, const float* __restrict__ Bi,
                                             int rsB, int csB,
                                             int tm, int tn, v8f& cre, v8f& cim) {
  const int lane = threadIdx.x & 31;
  const int mr = tm + (lane & 15);
  const int nc = tn + (lane & 15);
  const int kh = (lane >> 4) << 1;
  const float* Arp = Ar + (size_t)mr * rsA + (size_t)kh * csA;
  const float* Aip = Ai + (size_t)mr * rsA + (size_t)kh * csA;
  const float* Brp = Br + (size_t)kh * rsB + (size_t)nc * csB;
  const float* Bip = Bi + (size_t)kh * rsB + (size_t)nc * csB;
  v8f cn = {};
  #pragma unroll
  for (int k = 0; k < KPAD; k += 4) {
    v2f ar, ai, br, bi;
    ar.x = Arp[0];  ar.y = Arp[csA];
    ai.x = Aip[0];  ai.y = Aip[csA];
    br.x = Brp[0];  br.y = Brp[rsB];
    bi.x = Bip[0];  bi.y = Bip[rsB];
    cre = wmma4(ar, br, cre);
    cn  = wmma4(ai, bi, cn);
    cim = wmma4(ar, bi, cim);
    cim = wmma4(ai, br, cim);
    Arp += (size_t)4 * csA;  Aip += (size_t)4 * csA;
    Brp += (size_t)4 * rsB;  Bip += (size_t)4 * rsB;
  }
  cre = cre - cn;
}

// ---------------------------------------------------------------------------
// Input embedding: x(16,5,128,128) + coord grids -> project (7->64) with p_w,
// write into zero-padded 137x137 domain, layout [b][d][x][y].
// ---------------------------------------------------------------------------
__global__ void k_embed(const float* __restrict__ x, const float* __restrict__ p_w,
                        const float* __restrict__ p_b, float* __restrict__ X0) {
  size_t idx = (size_t)blockIdx.x * blockDim.x + threadIdx.x;
  const size_t total = (size_t)Bn * Cn * HW;
  if (idx >= total) return;
  int y = (int)(idx % Hh);
  size_t t = idx / Hh;
  int xr = (int)(t % Hh); t /= Hh;
  int d = (int)(t % Cn);
  int b = (int)(t / Cn);
  float v = 0.f;
  if (xr < Sn && y < Sn) {
    float acc = p_b[d];
    const float* xb = x + (size_t)b * 5 * Sn * Sn + (size_t)xr * Sn + y;
    #pragma unroll
    for (int c = 0; c < 5; ++c) acc += xb[(size_t)c * Sn * Sn] * p_w[c * Cn + d];
    acc += (xr * (1.0f / 127.0f)) * p_w[5 * Cn + d];
    acc += (y  * (1.0f / 127.0f)) * p_w[6 * Cn + d];
    v = acc;
  }
  X0[idx] = v;
}

// ---------------------------------------------------------------------------
// Twiddle tables, zero-padded to WMMA-friendly shapes:
//  Ey  [140][32] : n<16: cos(2pi ky y/137), n>=16: -sin. rows>=137 zero.
//  ExT [32][140] : cos / -sin of 2pi fx x/137. cols>=137 zero.
//  EIx [144][32] : cos / +sin. rows>=137 zero.
//  EIy [32][144] : c(ky)/N^2*cos ; rows 16..31: -c(ky)/N^2*sin. cols>=137 zero.
// fx(slot) = slot<16 ? slot : 121+(slot-16)
// ---------------------------------------------------------------------------
__global__ void k_twiddles(float* Ey, float* ExTr, float* ExTi,
                           float* EIxr, float* EIxi, float* EIy) {
  int i = blockIdx.x * blockDim.x + threadIdx.x;
  if (i >= 4608) return;
  const float TP = 6.28318530717958647692f;
  if (i < 4480) {
    // Ey [140][32]
    int y = i >> 5, n = i & 31, ky = n & 15;
    float v = 0.f;
    if (y < Hh) {
      int r = (ky * y) % Hh;
      float th = TP * (float)r / (float)Hh;
      v = (n < 16) ? cosf(th) : -sinf(th);
    }
    Ey[i] = v;
    // ExT [32][140]
    int s = i / 140, xx = i % 140;
    float er = 0.f, ei = 0.f;
    if (xx < Hh) {
      int fx = (s < 16) ? s : 121 + (s - 16);
      int rx = (fx * xx) % Hh;
      float tx = TP * (float)rx / (float)Hh;
      er = cosf(tx);
      ei = -sinf(tx);
    }
    ExTr[i] = er;
    ExTi[i] = ei;
  }
  {
    // EIx [144][32]
    int xx = i >> 5, s = i & 31;
    float cr = 0.f, ci = 0.f;
    if (xx < Hh) {
      int fx = (s < 16) ? s : 121 + (s - 16);
      int rx = (fx * xx) % Hh;
      float tx = TP * (float)rx / (float)Hh;
      cr = cosf(tx);
      ci = sinf(tx);
    }
    EIxr[i] = cr;
    EIxi[i] = ci;
  }
  {
    // EIy [32][144]
    int k = i / 144, yy = i % 144;
    float v = 0.f;
    if (yy < Hh) {
      int ky = k & 15;
      float cy = (ky == 0) ? 1.0f : 2.0f;
      float nrm = cy / ((float)Hh * (float)Hh);
      int ry = (ky * yy) % Hh;
      float ty = TP * (float)ry / (float)Hh;
      v = (k < 16) ? nrm * cosf(ty) : -nrm * sinf(ty);
    }
    EIy[i] = v;
  }
}

// ---------------------------------------------------------------------------
// Generic batched real GEMM: C[b] = act(A[b](MxK)*B[b](KxN) + bias + addend).
// 4 waves/block, one 16x16 tile per wave. Loads unguarded (see header note).
// ---------------------------------------------------------------------------
template<int KPAD>
__global__ void k_gemm(const float* __restrict__ A, int rsA, int csA, long sA,
                       const float* __restrict__ B, int rsB, int csB, long sB,
                       float* __restrict__ C, int ldc, long sC,
                       const float* __restrict__ bias,
                       const float* __restrict__ addend, long sAdd,
                       int M, int N, int act) {
  const int batch = blockIdx.y;
  const int Ntiles = (N + 15) >> 4;
  const int Mtiles = (M + 15) >> 4;
  const int tile = blockIdx.x * (blockDim.x >> 5) + (threadIdx.x >> 5);
  if (tile >= Mtiles * Ntiles) return;
  const int tm = (tile / Ntiles) << 4;
  const int tn = (tile % Ntiles) << 4;
  v8f acc = {};
  gemm_tile_u<KPAD>(A + (size_t)batch * sA, rsA, csA,
                    B + (size_t)batch * sB, rsB, csB, tm, tn, acc);
  const int lane = threadIdx.x & 31;
  const int col = tn + (lane & 15);
  if (col >= N) return;
  const int rb = tm + ((lane & 16) ? 8 : 0);
  float* Cb = C + (size_t)batch * sC;
  const float* Adb = addend ? addend + (size_t)batch * sAdd : nullptr;
  #pragma unroll
  for (int v = 0; v < 8; ++v) {
    const int r = rb + v;
    if (r < M) {
      float val = acc[v];
      if (bias) val += bias[r];
      if (Adb)  val += Adb[(size_t)r * ldc + col];
      if (act)  val = gelu_f(val);
      Cb[(size_t)r * ldc + col] = val;
    }
  }
}

// ---------------------------------------------------------------------------
// Forward x-DFT. One block per (b,c); 2 waves = 2 kx tiles.
// Fm[kx,ky] = sum_x ExT[kx,x] * F1[x,ky]  (complex x complex)
// F1: [(bc*137+x)*32 + ky(re 0..15/im 16..31)]; Fm: [(bc*32+kx)*16+ky] re/im.
// ---------------------------------------------------------------------------
__global__ void k_dftx(const float* __restrict__ FQ,
                       const float* __restrict__ ExTr, const float* __restrict__ ExTi,
                       float* __restrict__ Fmr, float* __restrict__ Fmi) {
  const int bc = blockIdx.x;                 // 0..1023
  const int wave = threadIdx.x >> 5;         // 0..1
  const float* Br = FQ + (size_t)bc * Hh * 32;
  v8f cre = {}, cim = {};
  const int tm = wave << 4;
  cgemm_tile_u<140>(ExTr, ExTi, 140, 1, Br, Br + 16, 32, 1, tm, 0, cre, cim);
  const int lane = threadIdx.x & 31;
  const int ky = lane & 15;
  const int rb = tm + ((lane & 16) ? 8 : 0);
  #pragma unroll
  for (int v = 0; v < 8; ++v) {
    const int kx = rb + v;
    const size_t off = ((size_t)bc * NKX + kx) * Mm + ky;
    Fmr[off] = cre[v];
    Fmi[off] = cim[v];
  }
}

// ---------------------------------------------------------------------------
// Spectral mixing. One block per mode (slot*16+ky); 4 waves = 4 o-tiles.
// G[b,o] = sum_i F[b,i]*w[i,o]  (complex 16x64x64). sw: [d][reim][i][o][kx][ky]
// ---------------------------------------------------------------------------
__global__ void k_specmul(const float* __restrict__ Fmr, const float* __restrict__ Fmi,
                          const float* __restrict__ sw1, const float* __restrict__ sw2,
                          int d, float* __restrict__ Gmr, float* __restrict__ Gmi) {
  const int mode = blockIdx.x;               // 0..511
  const int slot = mode >> 4, ky = mode & 15;
  const int wave = threadIdx.x >> 5;         // 0..3
  const float* sw = (slot < 16) ? sw1 : sw2;
  const int kxi = (slot < 16) ? slot : slot - 16;
  const size_t wbase = (size_t)d * 2 * Cn * Cn * 256 + (size_t)kxi * 16 + ky;
  const float* Br = sw + wbase;
  const float* Bi = sw + wbase + (size_t)Cn * Cn * 256;
  v8f cre = {}, cim = {};
  cgemm_tile_u<64>(Fmr + mode, Fmi + mode, Cn * NMODE, NMODE,
                   Br, Bi, Cn * 256, 256, 0, wave << 4, cre, cim);
  const int lane = threadIdx.x & 31;
  const int o = (wave << 4) + (lane & 15);
  const int rb = (lane & 16) ? 8 : 0;
  #pragma unroll
  for (int v = 0; v < 8; ++v) {
    const int b = rb + v;
    const size_t off = ((size_t)b * Cn + o) * NMODE + mode;
    Gmr[off] = cre[v];
    Gmi[off] = cim[v];
  }
}

// ---------------------------------------------------------------------------
// Inverse x-DFT. One block per (b,o); 4 waves loop over 9 x-tiles.
// S1[x,ky] = sum_slot EIx[x,slot]*G[slot,ky] -> FQ (re 0..15/im 16..31)
// ---------------------------------------------------------------------------
__global__ void k_idftx(const float* __restrict__ EIxr, const float* __restrict__ EIxi,
                        const float* __restrict__ Gmr, const float* __restrict__ Gmi,
                        float* __restrict__ FQ) {
  const int bc = blockIdx.x;                 // (b*64+o)
  const int wave = threadIdx.x >> 5;
  const float* Br = Gmr + (size_t)bc * NMODE;
  const float* Bi = Gmi + (size_t)bc * NMODE;
  const int lane = threadIdx.x & 31;
  const int ky = lane & 15;
  for (int t = wave; t < 9; t += 4) {
    v8f cre = {}, cim = {};
    const int tm = t << 4;
    cgemm_tile_u<32>(EIxr, EIxi, 32, 1, Br, Bi, 16, 1, tm, 0, cre, cim);
    const int rb = tm + ((lane & 16) ? 8 : 0);
    #pragma unroll
    for (int v = 0; v < 8; ++v) {
      const int xx = rb + v;
      if (xx < Hh) {
        const size_t off = ((size_t)bc * Hh + xx) * 32 + ky;
        FQ[off] = cre[v];
        FQ[off + 16] = cim[v];
      }
    }
  }
}

// ---------------------------------------------------------------------------
// Head: crop to 128x128, q1 (64->256) + GELU + q2 (256->1), fused per pixel.
// ---------------------------------------------------------------------------
__global__ void k_head(const float* __restrict__ X, const float* __restrict__ q1w,
                       const float* __restrict__ q1b, const float* __restrict__ q2w,
                       const float* __restrict__ q2b, float* __restrict__ out) {
  const int idx = blockIdx.x * blockDim.x + threadIdx.x;
  if (idx >= Bn * Sn * Sn) return;
  const int s = idx & (Sn * Sn - 1);
  const int b = idx >> 14;
  const int xr = s >> 7, y = s & 127;
  float xin[Cn];
  const float* Xb = X + (size_t)b * Cn * HW + (size_t)xr * Hh + y;
  #pragma unroll 8
  for (int c = 0; c < Cn; ++c) xin[c] = Xb[(size_t)c * HW];
  float acc = q2b[0];
  for (int h = 0; h < 256; ++h) {
    float t = q1b[h];
    const float* wr = q1w + h * Cn;
    #pragma unroll 8
    for (int c = 0; c < Cn; ++c) t += wr[c] * xin[c];
    acc += q2w[h] * gelu_f(t);
  }
  out[idx] = acc;
}

// ---------------------------------------------------------------------------
extern "C" void kernel_launch(void* const* d_in, const int* in_sizes, int n_in,
                              void* d_out, int out_size, void* d_ws, size_t ws_size,
                              hipStream_t stream) {
  const float* x      = (const float*)d_in[0];
  const float* p_w    = (const float*)d_in[1];
  const float* p_b    = (const float*)d_in[2];
  const float* sw1    = (const float*)d_in[3];
  const float* sw2    = (const float*)d_in[4];
  const float* mlp1_w = (const float*)d_in[5];
  const float* mlp1_b = (const float*)d_in[6];
  const float* mlp2_w = (const float*)d_in[7];
  const float* mlp2_b = (const float*)d_in[8];
  const float* ww     = (const float*)d_in[9];
  const float* wb     = (const float*)d_in[10];
  const float* q1_w   = (const float*)d_in[11];
  const float* q1_b   = (const float*)d_in[12];
  const float* q2_w   = (const float*)d_in[13];
  const float* q2_b   = (const float*)d_in[14];
  float* out = (float*)d_out;

  // Workspace carve-up (floats). Total ~64.3M floats (~257 MB).
  float* w0 = (float*)d_ws;
  const size_t SP = (size_t)Bn * Cn * HW;        // 19,219,456
  const size_t FQS = (size_t)RY * 32;            //  4,489,216
  const size_t MS = (size_t)Bn * Cn * NMODE;     //    524,288
  float* sp0 = w0;
  float* sp1 = sp0 + SP;
  float* sp2 = sp1 + SP;
  float* FQ  = sp2 + SP;
  float* Fmr = FQ + FQS;
  float* Fmi = Fmr + MS;
  float* Gmr = Fmi + MS;
  float* Gmi = Gmr + MS;
  float* Ey   = Gmi + MS;        // [140][32] = 4480
  float* ExTr = Ey + 4480;       // [32][140] = 4480
  float* ExTi = ExTr + 4480;
  float* EIxr = ExTi + 4480;     // [144][32] = 4608
  float* EIxi = EIxr + 4608;
  float* EIy  = EIxi + 4608;     // [32][144] = 4608

  k_twiddles<<<18, 256, 0, stream>>>(Ey, ExTr, ExTi, EIxr, EIxi, EIy);

  k_embed<<<(unsigned)((SP + 255) / 256), 256, 0, stream>>>(x, p_w, p_b, sp0);

  float* P = sp0;     // current activation
  float* S = sp1;     // spectral/mlp scratch
  float* T = sp2;     // scratch / next activation
  const long CHW = (long)Cn * HW;                // per-batch activation stride

  for (int d = 0; d < 4; ++d) {
    // 1) forward y-DFT: FQ[row,0:32] = P(row,137) * Ey(140,32); K padded to 140
    {
      int tiles = (RY / 16) * 2;
      k_gemm<140><<<dim3((tiles + 3) / 4, 1), 128, 0, stream>>>(
          P, Hh, 1, 0, Ey, 32, 1, 0, FQ, 32, 0,
          nullptr, nullptr, 0, RY, 32, 0);
    }
    // 2) forward x-DFT (complex) per (b,c)
    k_dftx<<<Bn * Cn, 64, 0, stream>>>(FQ, ExTr, ExTi, Fmr, Fmi);
    // 3) spectral mixing per mode (complex 16x64x64)
    k_specmul<<<NMODE, 128, 0, stream>>>(Fmr, Fmi, sw1, sw2, d, Gmr, Gmi);
    // 4) inverse x-DFT (complex) per (b,o) -> FQ reused as S1
    k_idftx<<<Bn * Cn, 128, 0, stream>>>(EIxr, EIxi, Gmr, Gmi, FQ);
    // 5) inverse y-DFT + Re(): S[row,0:137] = FQ(row,32) * EIy(32,144)
    {
      int tiles = (RY / 16) * 9;
      k_gemm<32><<<dim3((tiles + 3) / 4, 1), 128, 0, stream>>>(
          FQ, 32, 1, 0, EIy, 144, 1, 0, S, Hh, 0,
          nullptr, nullptr, 0, RY, Hh, 0);
    }
    // 6) T = gelu(mlp1 * S)
    k_gemm<64><<<dim3(1174, Bn), 128, 0, stream>>>(
        mlp1_w + (size_t)d * Cn * Cn, Cn, 1, 0, S, HW, 1, CHW, T, HW, CHW,
        mlp1_b + d * Cn, nullptr, 0, Cn, HW, 1);
    // 7) S = mlp2 * T
    k_gemm<64><<<dim3(1174, Bn), 128, 0, stream>>>(
        mlp2_w + (size_t)d * Cn * Cn, Cn, 1, 0, T, HW, 1, CHW, S, HW, CHW,
        mlp2_b + d * Cn, nullptr, 0, Cn, HW, 0);
    // 8) T = gelu(ww * P + S)   (residual path + combine)
    k_gemm<64><<<dim3(1174, Bn), 128, 0, stream>>>(
        ww + (size_t)d * Cn * Cn, Cn, 1, 0, P, HW, 1, CHW, T, HW, CHW,
        wb + d * Cn, S, CHW, Cn, HW, 1);
    // rotate: new activation is T
    float* tmp = P; P = T; T = tmp;
  }

  k_head<<<(Bn * Sn * Sn + 255) / 256, 256, 0, stream>>>(P, q1_w, q1_b, q2_w, q2_b, out);
}